// LoRAEmbedder_50766513438782
// MI455X (gfx1250) — compile-verified
//
#include <hip/hip_runtime.h>
#include <hip/hip_bf16.h>

typedef __attribute__((ext_vector_type(2))) float v2f;
typedef __attribute__((ext_vector_type(8))) float v8f;

#define TPB 256   // k1/k2 block size
#define TPB3 128  // k3 block size (4 waves)

// ---------------------------------------------------------------------------
// Kernel 1: h1[row][r] = sum_i x[row][i] * A[n][r][i]   (R = 16)
// ---------------------------------------------------------------------------
__global__ void lora_k1(const float* __restrict__ x, const float* __restrict__ A,
                        float* __restrict__ h1, int din) {
  const int row = blockIdx.x;
  const int n = row >> 3;
  const int tid = threadIdx.x;
  const float* xr = x + (size_t)row * din;
  const float* An = A + (size_t)n * 16 * din;
  float acc[16];
#pragma unroll
  for (int r = 0; r < 16; ++r) acc[r] = 0.f;
  for (int i = tid; i < din; i += TPB) {
    const float xv = xr[i];
#pragma unroll
    for (int r = 0; r < 16; ++r) acc[r] += xv * An[(size_t)r * din + i];
  }
  __shared__ float red[TPB];
#pragma unroll 1
  for (int r = 0; r < 16; ++r) {
    red[tid] = acc[r];
    __syncthreads();
    for (int off = TPB / 2; off > 0; off >>= 1) {
      if (tid < off) red[tid] += red[tid + off];
      __syncthreads();
    }
    if (tid == 0) h1[row * 16 + r] = red[0];
    __syncthreads();
  }
}

// ---------------------------------------------------------------------------
// Kernel 2: H[row][d] = sum_r h1[row][r] * B[n][d][r]  + per-row mean / rsig.
// ---------------------------------------------------------------------------
__global__ void lora_k2(const float* __restrict__ B, const float* __restrict__ h1,
                        float* __restrict__ H, float* __restrict__ mu,
                        float* __restrict__ rsig, int dout) {
  const int row = blockIdx.x;
  const int n = row >> 3;
  const int tid = threadIdx.x;
  __shared__ float sh1[16];
  if (tid < 16) sh1[tid] = h1[row * 16 + tid];
  __syncthreads();
  const float4* Bn = (const float4*)(B + (size_t)n * dout * 16);
  float* Hr = H + (size_t)row * dout;
  float s = 0.f, s2 = 0.f;
  for (int d = tid; d < dout; d += TPB) {
    const float4 b0 = Bn[d * 4 + 0];
    const float4 b1 = Bn[d * 4 + 1];
    const float4 b2 = Bn[d * 4 + 2];
    const float4 b3 = Bn[d * 4 + 3];
    float v = sh1[0] * b0.x + sh1[1] * b0.y + sh1[2] * b0.z + sh1[3] * b0.w
            + sh1[4] * b1.x + sh1[5] * b1.y + sh1[6] * b1.z + sh1[7] * b1.w
            + sh1[8] * b2.x + sh1[9] * b2.y + sh1[10] * b2.z + sh1[11] * b2.w
            + sh1[12] * b3.x + sh1[13] * b3.y + sh1[14] * b3.z + sh1[15] * b3.w;
    Hr[d] = v;
    s += v;
    s2 += v * v;
  }
  __shared__ float rs[TPB], rq[TPB];
  rs[tid] = s; rq[tid] = s2;
  __syncthreads();
  for (int off = TPB / 2; off > 0; off >>= 1) {
    if (tid < off) { rs[tid] += rs[tid + off]; rq[tid] += rq[tid + off]; }
    __syncthreads();
  }
  if (tid == 0) {
    const float inv = 1.f / (float)dout;
    const float m = rs[0] * inv;
    const float var = rq[0] * inv - m * m;
    mu[row] = m;
    rsig[row] = rsqrtf(var + 1e-5f);
  }
}

// ---------------------------------------------------------------------------
// Kernel 3: out[outrow][col] = sum_d Hn[m][d] * W[col][d] + c[col]
// WMMA f32 16x16x4, double-buffered LDS. W tiles stream via
// GLOBAL_LOAD_ASYNC_TO_LDS_B128 (ASYNCcnt); normalized H tiles go through the
// VGPR path (layernorm applied on the fly). 4 waves: wave w owns 16 columns.
// grid.x = 32 column strips (64 cols), grid.y = row-tile groups (MT tiles).
// ---------------------------------------------------------------------------
template <int MT>
__global__ void lora_k3(const float* __restrict__ H, const float* __restrict__ W,
                        const float* __restrict__ g, const float* __restrict__ be,
                        const float* __restrict__ c, const float* __restrict__ mu,
                        const float* __restrict__ rsig, float* __restrict__ out,
                        int dout, int M, int rowBase, int rowStride) {
  constexpr int KC = 32;
  constexpr int HSTR = KC + 2;  // 34: bank-conflict pad, keeps 8B align for b64 reads
  constexpr int WSTR = KC + 4;  // 36: rows 144B => 16B-aligned async B128 stores
  constexpr int NCOL = 64;      // columns per workgroup
  __shared__ __align__(16) float sH[2][MT * 16 * HSTR];
  __shared__ __align__(16) float sW[2][NCOL * WSTR];

  const int tid = threadIdx.x;
  const int wave = tid >> 5;
  const int lane = tid & 31;
  const int half = lane >> 4;
  const int l16 = lane & 15;
  const int col0 = blockIdx.x * NCOL;
  const int m0 = blockIdx.y * MT * 16;

  v8f acc[MT];
#pragma unroll
  for (int t = 0; t < MT; ++t)
#pragma unroll
    for (int j = 0; j < 8; ++j) acc[t][j] = 0.f;

  // ---- staging helpers -----------------------------------------------------
  auto stageW = [&](int buf, int k0) {
    // NCOL rows x KC floats, 16B per lane per async copy (fully coalesced along dout)
#pragma unroll
    for (int it = 0; it < (NCOL * (KC / 4)) / TPB3; ++it) {
      const int idx = it * TPB3 + tid;
      const int r = idx >> 3;        // local column (row of W)
      const int seg = idx & 7;       // 16B segment within the KC chunk
      const float* gp = W + (size_t)(col0 + r) * dout + k0 + seg * 4;
      const unsigned ldsOff = (unsigned)(size_t)&sW[buf][r * WSTR + seg * 4];
      asm volatile("global_load_async_to_lds_b128 %0, %1, off"
                   :: "v"(ldsOff), "v"((unsigned long long)(size_t)gp)
                   : "memory");
    }
  };
  auto stageH = [&](int buf, int k0) {
#pragma unroll
    for (int it = 0; it < (MT * 16 * (KC / 4)) / TPB3; ++it) {
      const int idx = it * TPB3 + tid;
      const int r = idx / (KC / 4);
      const int seg = idx % (KC / 4);
      const int m = m0 + r;
      const int k = k0 + seg * 4;
      float4 hv = make_float4(0.f, 0.f, 0.f, 0.f);
      float mur = 0.f, rr = 0.f;
      if (m < M) {
        hv = *(const float4*)(H + (size_t)m * dout + k);
        mur = mu[m];
        rr = rsig[m];
      }
      const float4 gv = *(const float4*)(g + k);
      const float4 bv = *(const float4*)(be + k);
      float* dst = &sH[buf][r * HSTR + seg * 4];
      dst[0] = (hv.x - mur) * rr * gv.x + bv.x;
      dst[1] = (hv.y - mur) * rr * gv.y + bv.y;
      dst[2] = (hv.z - mur) * rr * gv.z + bv.z;
      dst[3] = (hv.w - mur) * rr * gv.w + bv.w;
    }
  };

  const int nchunks = dout / KC;  // all dout are multiples of 32

  // prologue: stage chunk 0 into buffer 0
  stageW(0, 0);
  stageH(0, 0);
  asm volatile("s_wait_asynccnt 0" ::: "memory");
  __syncthreads();

  for (int ci = 0; ci < nchunks; ++ci) {
    const int buf = ci & 1;
    // kick off next chunk's staging into the other buffer (overlaps with WMMA)
    if (ci + 1 < nchunks) {
      stageW(buf ^ 1, (ci + 1) * KC);
      stageH(buf ^ 1, (ci + 1) * KC);
    }
    // ---- WMMA over current chunk ----
#pragma unroll
    for (int kb = 0; kb < KC; kb += 4) {
      const float* wrow = &sW[buf][(wave * 16 + l16) * WSTR + kb];
      v2f b;
      b[0] = wrow[half];
      b[1] = wrow[half + 2];
#pragma unroll
      for (int t = 0; t < MT; ++t) {
        const float* hrow = &sH[buf][(t * 16 + l16) * HSTR + kb + 2 * half];
        v2f a;
        a[0] = hrow[0];
        a[1] = hrow[1];
        acc[t] = __builtin_amdgcn_wmma_f32_16x16x4_f32(
            false, a, false, b, (short)0, acc[t], false, false);
      }
    }
    asm volatile("s_wait_asynccnt 0" ::: "memory");
    __syncthreads();
  }

  // ---- store with row scatter + bias (divergence only after all WMMAs) ----
  const int col = col0 + wave * 16 + l16;
  const float cv = c[col];
#pragma unroll
  for (int t = 0; t < MT; ++t) {
#pragma unroll
    for (int v = 0; v < 8; ++v) {
      const int m = m0 + t * 16 + v + 8 * half;
      if (m < M) {
        const int outrow = rowBase + (m >> 3) * rowStride + (m & 7);
        out[(size_t)outrow * 2048 + col] = acc[t][v] + cv;
      }
    }
  }
}

// ---------------------------------------------------------------------------
extern "C" void kernel_launch(void* const* d_in, const int* in_sizes, int n_in,
                              void* d_out, int out_size, void* d_ws, size_t ws_size,
                              hipStream_t stream) {
  struct Meta { int din, dout, n, rowBase, rowStride; };
  static const Meta metas[13] = {
      // DOUBLE (n=19): rowBase = t*8, stride 80
      {3072, 9216, 19, 0, 80},    {3072, 3072, 19, 8, 80},
      {3072, 12288, 19, 16, 80},  {12288, 3072, 19, 24, 80},
      {3072, 18432, 19, 32, 80},  {3072, 9216, 19, 40, 80},
      {3072, 3072, 19, 48, 80},   {3072, 12288, 19, 56, 80},
      {12288, 3072, 19, 64, 80},  {3072, 18432, 19, 72, 80},
      // SINGLE (n=38): rowBase = 1520 + t*8, stride 24
      {3072, 21504, 38, 1520, 24}, {15360, 3072, 38, 1528, 24},
      {3072, 9216, 38, 1536, 24},
  };

  float* Hraw = (float*)d_ws;            // max 304 * 21504 floats (~26.1 MB)
  float* h1 = Hraw + (size_t)304 * 21504;
  float* muP = h1 + 304 * 16;
  float* rsP = muP + 304;
  float* out = (float*)d_out;

  constexpr int MTILES = 2;
  for (int i = 0; i < 13; ++i) {
    const Meta& mt = metas[i];
    const float* A = (const float*)d_in[7 * i + 0];
    const float* B = (const float*)d_in[7 * i + 1];
    const float* x = (const float*)d_in[7 * i + 2];
    const float* g = (const float*)d_in[7 * i + 3];
    const float* be = (const float*)d_in[7 * i + 4];
    const float* W = (const float*)d_in[7 * i + 5];
    const float* c = (const float*)d_in[7 * i + 6];
    const int M = mt.n * 8;

    lora_k1<<<M, TPB, 0, stream>>>(x, A, h1, mt.din);
    lora_k2<<<M, TPB, 0, stream>>>(B, h1, Hraw, muP, rsP, mt.dout);

    const int Mt = (M + 15) / 16;
    dim3 grid3(2048 / 64, (Mt + MTILES - 1) / MTILES);
    lora_k3<MTILES><<<grid3, TPB3, 0, stream>>>(Hraw, W, g, be, c, muP, rsP, out,
                                                mt.dout, M, mt.rowBase, mt.rowStride);
  }
}